// EMA_19129784336961
// MI455X (gfx1250) — compile-verified
//
#include <hip/hip_runtime.h>
#include <math.h>

// Problem constants (from reference): B=8, C=256, GROUP=32 -> BG=256 groups, CG=8, H=W=128
#define BG   256
#define CG   8
#define H    128
#define W    128
#define HW   (H*W)      // 16384
#define EPSV 1e-5f

typedef float v8f __attribute__((ext_vector_type(8)));
typedef float v2f __attribute__((ext_vector_type(2)));

// Fast sigmoid: v_exp_f32 + v_rcp_f32 (avoid IEEE div_scale chain)
__device__ __forceinline__ float sigf(float v) {
    return __builtin_amdgcn_rcpf(1.f + __expf(-v));
}

// LDS byte offset of a generic pointer that points into LDS
__device__ __forceinline__ unsigned lds_off_u32(const void* p) {
    return (unsigned)(unsigned long long)(__attribute__((address_space(3))) const void*)p;
}

// ---------------------------------------------------------------------------
// Pass 1: per (bg,c): rowsum[y] = sum_x gx(y,x), colsum[x] = sum_y gx(y,x)
// grid = BG*CG blocks, 128 threads (thread = column)
// ---------------------------------------------------------------------------
__global__ void k_rowcol(const float* __restrict__ x,
                         float* __restrict__ rowsum, float* __restrict__ colsum) {
    const int blk = blockIdx.x;                 // (bg*8 + c)
    const int t   = threadIdx.x;                // column
    const int lane = t & 31, wave = t >> 5;     // wave32
    const float* p = x + (size_t)blk * HW;

    __shared__ float lpart[4][H];

    float colacc = 0.f;
    for (int y = 0; y < H; ++y) {
        if (y + 8 < H) __builtin_prefetch(p + (y + 8) * W + t, 0, 0);
        float v = p[y * W + t];
        colacc += v;
        #pragma unroll
        for (int off = 16; off > 0; off >>= 1) v += __shfl_down(v, off, 32);
        if (lane == 0) lpart[wave][y] = v;
    }
    __syncthreads();
    float rtot = lpart[0][t] + lpart[1][t] + lpart[2][t] + lpart[3][t];
    rowsum[blk * H + t] = rtot;
    colsum[blk * W + t] = colacc;
}

// ---------------------------------------------------------------------------
// Pass 2 (tiny, WMMA): hw = w1(8x8) @ [rowmean;colmean](8x256) + b1 -> sigmoid
// One block per group, 128 threads (4 waves), V_WMMA_F32_16X16X4_F32 x2 per tile
// ---------------------------------------------------------------------------
__global__ void k_gates(const float* __restrict__ rowsum, const float* __restrict__ colsum,
                        const float* __restrict__ w1, const float* __restrict__ b1,
                        float* __restrict__ sh, float* __restrict__ sw) {
    const int bg  = blockIdx.x;
    const int tid = threadIdx.x;
    __shared__ float m[CG * 256];

    #pragma unroll
    for (int i = 0; i < 16; ++i) {
        int idx = i * 128 + tid;               // < 2048
        int c = idx >> 8, s = idx & 255;
        float v = (s < H) ? rowsum[(bg * CG + c) * H + s]
                          : colsum[(bg * CG + c) * W + (s - H)];
        m[idx] = v * (1.f / 128.f);
    }
    __syncthreads();

    const int lane  = tid & 31;
    const int wave  = tid >> 5;
    const int mrow  = lane & 15;
    const int kbase = (lane >> 4) * 2;         // 0 or 2

    v2f a0, a1f;
    a0.x  = (mrow < CG) ? w1[mrow * CG + kbase + 0] : 0.f;
    a0.y  = (mrow < CG) ? w1[mrow * CG + kbase + 1] : 0.f;
    a1f.x = (mrow < CG) ? w1[mrow * CG + 4 + kbase + 0] : 0.f;
    a1f.y = (mrow < CG) ? w1[mrow * CG + 4 + kbase + 1] : 0.f;

    for (int t = wave; t < 16; t += 4) {       // 16 tiles of 16 positions
        int s = t * 16 + (lane & 15);
        v2f b0, b1v;
        b0.x  = m[(kbase + 0) * 256 + s];
        b0.y  = m[(kbase + 1) * 256 + s];
        b1v.x = m[(4 + kbase + 0) * 256 + s];
        b1v.y = m[(4 + kbase + 1) * 256 + s];

        v8f acc = {0.f, 0.f, 0.f, 0.f, 0.f, 0.f, 0.f, 0.f};
        acc = __builtin_amdgcn_wmma_f32_16x16x4_f32(false, a0,  false, b0,  (short)0, acc, false, false);
        acc = __builtin_amdgcn_wmma_f32_16x16x4_f32(false, a1f, false, b1v, (short)0, acc, false, false);

        if (lane < 16) {                       // lanes 0..15 hold rows M=0..7 in acc[0..7]
            int ss = t * 16 + lane;
            #pragma unroll
            for (int o = 0; o < CG; ++o) {
                float sg = sigf(acc[o] + b1[o]);
                if (ss < H) sh[(bg * CG + o) * H + ss] = sg;
                else        sw[(bg * CG + o) * W + (ss - H)] = sg;
            }
        }
    }
}

// ---------------------------------------------------------------------------
// Pass 3: per (bg,c): sum & sumsq of gated = gx * sh[y] * sw[x]
// ---------------------------------------------------------------------------
__global__ void k_gsum(const float* __restrict__ x,
                       const float* __restrict__ sh, const float* __restrict__ sw,
                       float* __restrict__ gsum, float* __restrict__ gsq) {
    const int blk = blockIdx.x;
    const int tid = threadIdx.x;
    const float* p   = x  + (size_t)blk * HW;
    const float* shp = sh + blk * H;
    const float* swp = sw + blk * W;

    float s = 0.f, s2 = 0.f;
    for (int i = 0; i < HW / 256; ++i) {
        int idx = i * 256 + tid;
        if (i + 4 < HW / 256) __builtin_prefetch(p + idx + 1024, 0, 0);
        int y = idx >> 7, xx = idx & 127;
        float v = p[idx] * shp[y] * swp[xx];
        s += v; s2 += v * v;
    }
    __shared__ float rs_[256], rq_[256];
    rs_[tid] = s; rq_[tid] = s2;
    __syncthreads();
    for (int off = 128; off > 0; off >>= 1) {
        if (tid < off) { rs_[tid] += rs_[tid + off]; rq_[tid] += rq_[tid + off]; }
        __syncthreads();
    }
    if (tid == 0) { gsum[blk] = rs_[0]; gsq[blk] = rq_[0]; }
}

// ---------------------------------------------------------------------------
// Pass 4 (tiny): per-group stats + collapsed weights (see derivation round 0)
// ---------------------------------------------------------------------------
__global__ void k_prep(const float* __restrict__ x,
                       const float* __restrict__ rowsum, const float* __restrict__ colsum,
                       const float* __restrict__ gsum, const float* __restrict__ gsq,
                       const float* __restrict__ w3, const float* __restrict__ b3,
                       const float* __restrict__ gamma, const float* __restrict__ beta,
                       float* __restrict__ a1, float* __restrict__ weff,
                       float* __restrict__ kconst) {
    const int bg = blockIdx.x;
    const int lane = threadIdx.x;
    __shared__ float S[CG][9], mu_s[CG], rs_s[CG], x1m[CG], m2[CG], x11s[CG], x21s[CG];

    if (lane < CG) {
        int c = lane;
        const float* rp = rowsum + (bg * CG + c) * H;
        const float* cp = colsum + (bg * CG + c) * W;
        float tot = 0.f;
        for (int y = 0; y < H; ++y) tot += rp[y];
        float r0 = rp[0], rN = rp[H - 1], c0 = cp[0], cN = cp[W - 1];
        const float* g = x + (size_t)(bg * CG + c) * HW;
        float g00 = g[0], g0N = g[W - 1], gN0 = g[(H - 1) * W], gNN = g[(H - 1) * W + W - 1];
        #pragma unroll
        for (int kh = 0; kh < 3; ++kh) {
            #pragma unroll
            for (int kw = 0; kw < 3; ++kw) {
                float R = (kh == 0) ? rN : ((kh == 2) ? r0 : 0.f);
                float Cc = (kw == 0) ? cN : ((kw == 2) ? c0 : 0.f);
                float X = 0.f;
                if (kh == 0 && kw == 0) X = gNN;
                if (kh == 0 && kw == 2) X = gN0;
                if (kh == 2 && kw == 0) X = g0N;
                if (kh == 2 && kw == 2) X = g00;
                S[c][kh * 3 + kw] = tot - R - Cc + X;
            }
        }
        float mu = gsum[bg * CG + c] * (1.f / (float)HW);
        float var = gsq[bg * CG + c] * (1.f / (float)HW) - mu * mu;
        mu_s[c] = mu;
        rs_s[c] = rsqrtf(var + EPSV);
        x1m[c]  = beta[c];   // mean of x1 over HW == gn_beta exactly
    }
    __syncthreads();
    if (lane < CG) {
        int c = lane;
        float acc = 0.f;
        #pragma unroll
        for (int c2 = 0; c2 < CG; ++c2)
            #pragma unroll
            for (int t = 0; t < 9; ++t)
                acc += w3[(c * CG + c2) * 9 + t] * S[c2][t];
        m2[c] = b3[c] + acc * (1.f / (float)HW);
    }
    __syncthreads();
    if (lane == 0) {
        float mx1 = x1m[0], mx2 = m2[0];
        for (int c = 1; c < CG; ++c) { mx1 = fmaxf(mx1, x1m[c]); mx2 = fmaxf(mx2, m2[c]); }
        float d1 = 0.f, d2 = 0.f;
        for (int c = 0; c < CG; ++c) {
            x11s[c] = expf(x1m[c] - mx1); d1 += x11s[c];
            x21s[c] = expf(m2[c]  - mx2); d2 += x21s[c];
        }
        float i1 = 1.f / d1, i2 = 1.f / d2;
        for (int c = 0; c < CG; ++c) { x11s[c] *= i1; x21s[c] *= i2; }
    }
    __syncthreads();
    if (lane < CG) {
        int c = lane;
        a1[bg * CG + c] = x21s[c] * rs_s[c] * gamma[c];
    }
    for (int t = lane; t < CG * 9; t += 32) {
        int c2 = t / 9, tap = t % 9;
        float acc = 0.f;
        #pragma unroll
        for (int c = 0; c < CG; ++c) acc += x11s[c] * w3[(c * CG + c2) * 9 + tap];
        weff[bg * CG * 9 + t] = acc;
    }
    if (lane == 0) {
        float k = 0.f;
        for (int c = 0; c < CG; ++c)
            k += x21s[c] * (beta[c] - mu_s[c] * rs_s[c] * gamma[c]) + x11s[c] * b3[c];
        kconst[bg] = k;
    }
}

// ---------------------------------------------------------------------------
// Pass 5: fused final. 16-row tile + 1-row halo per side, all 8 channels in LDS
// staged with GLOBAL_LOAD_ASYNC_TO_LDS_B128 (ASYNCcnt), then per-pixel scalar
//   wsum = kconst + sum_c a1[c]*sh[y]*sw[x]*gx + sum_{c,tap} weff*neighborhood
// out(c,p) = gx(c,p) * sigmoid(wsum).  grid = BG*8 blocks, 256 threads.
// ---------------------------------------------------------------------------
#define TILE_H 16
__global__ void k_final(const float* __restrict__ x,
                        const float* __restrict__ sh, const float* __restrict__ sw,
                        const float* __restrict__ a1, const float* __restrict__ weff,
                        const float* __restrict__ kconst,
                        float* __restrict__ out) {
    const int bg   = blockIdx.x >> 3;
    const int tile = blockIdx.x & 7;
    const int y0   = tile * TILE_H;
    const int tid  = threadIdx.x;

    __shared__ float tileS[CG * (TILE_H + 2) * W];  // 73728 B
    __shared__ float sws[CG * W];                   // 4 KB
    __shared__ float shs[CG * TILE_H];
    __shared__ float weffs[CG * 9];
    __shared__ float a1s[CG];
    __shared__ float kc_s;

    // stage 18 rows x 8 channels (4608 float4; 18 per thread) via async copy
    const float* gbase = x + (size_t)(bg * CG) * HW;
    #pragma unroll
    for (int i = 0; i < TILE_H + 2; ++i) {
        int flat = i * 256 + tid;                   // < 4608
        int c = flat / 576, rem = flat % 576;       // 18 rows * 32 q per channel
        int r = rem >> 5, q = rem & 31;             // row in [0,18), float4 col
        int y = y0 - 1 + r;
        float* dst = &tileS[(c * (TILE_H + 2) + r) * W + q * 4];
        if (y >= 0 && y < H) {
            unsigned long long ga =
                (unsigned long long)(gbase + (size_t)c * HW + (size_t)y * W + q * 4);
            unsigned lo = lds_off_u32(dst);
            asm volatile("global_load_async_to_lds_b128 %0, %1, off"
                         :: "v"(lo), "v"(ga) : "memory");
        } else {
            *(float4*)dst = make_float4(0.f, 0.f, 0.f, 0.f);
        }
    }
    if (tid < CG * TILE_H) {
        int c = tid >> 4, yl = tid & (TILE_H - 1);
        shs[tid] = sh[(bg * CG + c) * H + y0 + yl];
    }
    #pragma unroll
    for (int j = 0; j < 4; ++j) {
        int idx = j * 256 + tid;
        if (idx < CG * W) {
            int c = idx >> 7, xx = idx & 127;
            sws[idx] = sw[(bg * CG + c) * W + xx];
        }
    }
    if (tid < CG * 9) weffs[tid] = weff[bg * CG * 9 + tid];
    if (tid < CG)     a1s[tid]   = a1[bg * CG + tid];
    if (tid == 0)     kc_s       = kconst[bg];
    asm volatile("s_wait_asynccnt 0" ::: "memory");
    __syncthreads();

    const int xx   = tid & 127;
    const int half = tid >> 7;
    const bool xl = (xx > 0), xr = (xx < W - 1);

    for (int j = 0; j < TILE_H / 2; ++j) {
        int yl = half * (TILE_H / 2) + j;           // local output row
        float wsum = kc_s;
        #pragma unroll
        for (int c = 0; c < CG; ++c) {
            const float* rm = &tileS[(c * (TILE_H + 2) + yl) * W];  // y-1
            const float* rc = rm + W;                               // y
            const float* rp = rc + W;                               // y+1
            float cen = rc[xx];
            wsum += a1s[c] * shs[c * TILE_H + yl] * sws[c * W + xx] * cen;
            float tl = xl ? rm[xx - 1] : 0.f;
            float tr = xr ? rm[xx + 1] : 0.f;
            float cl = xl ? rc[xx - 1] : 0.f;
            float cr = xr ? rc[xx + 1] : 0.f;
            float bl = xl ? rp[xx - 1] : 0.f;
            float br = xr ? rp[xx + 1] : 0.f;
            const float* wv = &weffs[c * 9];
            wsum += wv[0] * tl + wv[1] * rm[xx] + wv[2] * tr
                  + wv[3] * cl + wv[4] * cen    + wv[5] * cr
                  + wv[6] * bl + wv[7] * rp[xx] + wv[8] * br;
        }
        float sg = sigf(wsum);
        int y = y0 + yl;
        size_t ob = (size_t)(bg * CG) * HW + (size_t)y * W + xx;
        #pragma unroll
        for (int c = 0; c < CG; ++c)
            out[ob + (size_t)c * HW] = tileS[(c * (TILE_H + 2) + yl + 1) * W + xx] * sg;
    }
}

// ---------------------------------------------------------------------------
extern "C" void kernel_launch(void* const* d_in, const int* in_sizes, int n_in,
                              void* d_out, int out_size, void* d_ws, size_t ws_size,
                              hipStream_t stream) {
    const float* x     = (const float*)d_in[0];
    const float* w1    = (const float*)d_in[1];
    const float* b1    = (const float*)d_in[2];
    const float* gamma = (const float*)d_in[3];
    const float* beta  = (const float*)d_in[4];
    const float* w3    = (const float*)d_in[5];
    const float* b3    = (const float*)d_in[6];
    float* out = (float*)d_out;

    float* ws = (float*)d_ws;
    float* rowsum = ws;                 ws += BG * CG * H;
    float* colsum = ws;                 ws += BG * CG * W;
    float* sh     = ws;                 ws += BG * CG * H;
    float* sw     = ws;                 ws += BG * CG * W;
    float* gsum   = ws;                 ws += BG * CG;
    float* gsq    = ws;                 ws += BG * CG;
    float* a1     = ws;                 ws += BG * CG;
    float* weff   = ws;                 ws += BG * CG * 9;
    float* kconst = ws;                 ws += BG;

    k_rowcol<<<BG * CG, 128, 0, stream>>>(x, rowsum, colsum);
    k_gates <<<BG,      128, 0, stream>>>(rowsum, colsum, w1, b1, sh, sw);
    k_gsum  <<<BG * CG, 256, 0, stream>>>(x, sh, sw, gsum, gsq);
    k_prep  <<<BG,       32, 0, stream>>>(x, rowsum, colsum, gsum, gsq, w3, b3,
                                          gamma, beta, a1, weff, kconst);
    k_final <<<BG * 8, 256, 0, stream>>>(x, sh, sw, a1, weff, kconst, out);
}